// DTCWT1D_28673201668336
// MI455X (gfx1250) — compile-verified
//
#include <hip/hip_runtime.h>

typedef float v2f __attribute__((ext_vector_type(2)));
typedef float v4f __attribute__((ext_vector_type(4)));
typedef float v8f __attribute__((ext_vector_type(8)));

static constexpr int B  = 64;
static constexpr int L  = 1 << 20;     // 1048576
static constexpr int L1 = L >> 1;      // 524288
static constexpr int L2 = L >> 2;      // 262144
static constexpr int L3 = L >> 3;      // 131072

static constexpr int N3  = 512;        // level-3 outputs per workgroup
static constexpr int GPB = L3 / N3;    // 256 groups per batch row
static constexpr int T1  = 134;        // level-1 tiles, base1-48 .. base1+2095
static constexpr int T2  = 66;         // level-2 tiles, base2-16 .. base2+1039
static constexpr int T3  = 32;         // level-3 tiles, base3    .. base3+511

static constexpr int SX  = 4304;       // staged x (4294 used), base = X0-99
static constexpr int SL1 = 2176;       // lo1 (2144 used), base = base1-48
static constexpr int SL2 = 1088;       // lo2 (1056 used), base = base2-16

// Level-1 combined filter table, both filters referenced to pad=3.
__device__ __forceinline__ float l1tap(int j, int k, const float* h0o, const float* h1o) {
  if (j == 0) return (k >= 1 && k <= 5) ? h0o[k - 1] : 0.f;
  if (j == 1) return (k <= 6) ? h1o[k] : 0.f;
  return 0.f;
}
// Levels 2/3: columns j = {h0a, h1a, h1b}, 14 taps, pad=6 (k padded to 16)
__device__ __forceinline__ float l2tap(int j, int k, const float* h0a,
                                       const float* h1a, const float* h1b) {
  if (k >= 14) return 0.f;
  if (j == 0) return h0a[k];
  if (j == 1) return h1a[k];
  if (j == 2) return h1b[k];
  return 0.f;
}

__device__ __forceinline__ void nt_store4(float* p, float a, float b2, float c, float d) {
  v4f v = {a, b2, c, d};
  __builtin_nontemporal_store(v, (v4f*)p);
}

__device__ __forceinline__ void lds_store8(float* base, const v8f& acc) {
  v4f* p = (v4f*)base;
  p[0] = (v4f){acc[0], acc[1], acc[2], acc[3]};
  p[1] = (v4f){acc[4], acc[5], acc[6], acc[7]};
}

__device__ __forceinline__ void lds_store8_masked(float* base, const v8f& acc,
                                                  int nb, int lim) {
  float v[8];
  for (int d = 0; d < 8; ++d) {
    int n = nb + d;
    v[d] = (n >= 0 && n < lim) ? acc[d] : 0.f;
  }
  v4f* p = (v4f*)base;
  p[0] = (v4f){v[0], v[1], v[2], v[3]};
  p[1] = (v4f){v[4], v[5], v[6], v[7]};
}

__global__ __launch_bounds__(256)
void dtcwt1d_fused(const float* __restrict__ x,
                   const float* __restrict__ h0o, const float* __restrict__ h1o,
                   const float* __restrict__ h0a, const float* __restrict__ h1a,
                   const float* __restrict__ h1b,
                   float* __restrict__ o_lo,  float* __restrict__ o_yh0,
                   float* __restrict__ o_yh1, float* __restrict__ o_yh2) {
  __shared__ float s_x[SX];
  __shared__ float s_lo1[SL1];
  __shared__ float s_lo2[SL2];

  const int b     = blockIdx.x / GPB;
  const int g     = blockIdx.x % GPB;
  const int base3 = g * N3;
  const int base2 = base3 * 2;
  const int base1 = base2 * 2;
  const int X0    = base1 * 2;
  const bool edge = (g == 0) || (g == GPB - 1);   // only blocks that touch SAME pads

  const int tid  = threadIdx.x;
  const int lane = tid & 31;
  // force wave id scalar: tile loops + ownership guards become SALU branches
  const int wave = __builtin_amdgcn_readfirstlane(tid) >> 5;
  const int j    = lane & 15;   // filter column (and A row index for loads)
  const int half = lane >> 4;   // K-half / M-half selector

  // ---- stage x tile (+halo) into LDS with SAME zero padding ----
  const int XLO = X0 - 99;
  const int NX  = 4294;                           // max x index = X0 + 4194
  const float* xb = x + (size_t)b * L;
  for (int idx = tid; idx < NX; idx += 256) {
    int gix = XLO + idx;
    s_x[idx] = (gix >= 0 && gix < L) ? __builtin_nontemporal_load(&xb[gix]) : 0.f;
  }

  // ---- per-lane B operands (filter matrices), one v2f per K-chunk of 4 ----
  // B 4x16: lanes 0-15 hold K={0,1} (V0,V1), lanes 16-31 hold K={2,3}; N = lane&15
  v2f b1[2], b2[4];
  for (int c = 0; c < 2; ++c) {
    int k0 = 4 * c + 2 * half;
    b1[c].x = l1tap(j, k0,     h0o, h1o);
    b1[c].y = l1tap(j, k0 + 1, h0o, h1o);
  }
  for (int c = 0; c < 4; ++c) {
    int k0 = 4 * c + 2 * half;
    b2[c].x = l2tap(j, k0,     h0a, h1a, h1b);
    b2[c].y = l2tap(j, k0 + 1, h0a, h1a, h1b);
  }

  float* const yh0b = o_yh0 + (size_t)b * L1;
  float* const yh1a = o_yh1 + ((size_t)b * 2 + 0) * L2;
  float* const yh1b = o_yh1 + ((size_t)b * 2 + 1) * L2;
  float* const lo3b = o_lo  + (size_t)b * L3;
  float* const yh2a = o_yh2 + ((size_t)b * 2 + 0) * L3;
  float* const yh2b = o_yh2 + ((size_t)b * 2 + 1) * L3;

  __syncthreads();

  // ================= level 1: 2 chained K=4 f32 WMMAs per 16-output tile =================
  for (int t = wave; t < T1; t += 8) {
    const int n0   = base1 - 48 + 16 * t;          // scalar; n0 ≡ 0 (mod 16)
    const int arow = 2 * (n0 + j) + 96 - X0;       // s_x index of tap k=0 for row i=j
    v8f acc = {};
    for (int c = 0; c < 2; ++c) {
      int k0 = 4 * c + 2 * half;                   // A 16x4: half-split K, V0/V1 in half
      v2f a; a.x = s_x[arow + k0]; a.y = s_x[arow + k0 + 1];
      acc = __builtin_amdgcn_wmma_f32_16x16x4_f32(false, a, false, b1[c],
                                                  (short)0, acc, false, false);
    }
    const int nb = n0 + 8 * half;                  // D: vgpr d -> output n0 + 8*half + d
    if (j == 0) {                                  // lowpass -> LDS, zero outside [0,L1)
      float* dst = &s_lo1[nb - (base1 - 48)];
      if (edge) lds_store8_masked(dst, acc, nb, L1);
      else      lds_store8(dst, acc);
    } else if (j == 1 && n0 >= base1 && n0 < base1 + 2048) {  // scalar tile guard
      nt_store4(yh0b + nb,     acc[0], acc[1], acc[2], acc[3]);
      nt_store4(yh0b + nb + 4, acc[4], acc[5], acc[6], acc[7]);
    }
  }

  __syncthreads();

  // ================= level 2: 4 chained K=4 WMMAs, filters {h0a,h1a,h1b} =================
  for (int t = wave; t < T2; t += 8) {
    const int n0   = base2 - 16 + 16 * t;          // scalar, aligned
    const int arow = 2 * (n0 + j) + 42 - base1;    // s_lo1 index of tap k=0 for row i=j
    v8f acc = {};
    for (int c = 0; c < 4; ++c) {
      int k0 = 4 * c + 2 * half;
      v2f a; a.x = s_lo1[arow + k0]; a.y = s_lo1[arow + k0 + 1];
      acc = __builtin_amdgcn_wmma_f32_16x16x4_f32(false, a, false, b2[c],
                                                  (short)0, acc, false, false);
    }
    const int nb = n0 + 8 * half;
    const bool own = (n0 >= base2) && (n0 < base2 + 1024);     // scalar per tile
    if (j == 0) {
      float* dst = &s_lo2[nb - (base2 - 16)];
      if (edge) lds_store8_masked(dst, acc, nb, L2);
      else      lds_store8(dst, acc);
    } else if (j == 1 && own) {
      nt_store4(yh1a + nb,     acc[0], acc[1], acc[2], acc[3]);
      nt_store4(yh1a + nb + 4, acc[4], acc[5], acc[6], acc[7]);
    } else if (j == 2 && own) {
      nt_store4(yh1b + nb,     acc[0], acc[1], acc[2], acc[3]);
      nt_store4(yh1b + nb + 4, acc[4], acc[5], acc[6], acc[7]);
    }
  }

  __syncthreads();

  // ================= level 3: all outputs in-range, no guards =================
  for (int t = wave; t < T3; t += 8) {
    const int n0   = base3 + 16 * t;
    const int arow = 2 * (n0 + j) + 10 - base2;    // s_lo2 index of tap k=0 for row i=j
    v8f acc = {};
    for (int c = 0; c < 4; ++c) {
      int k0 = 4 * c + 2 * half;
      v2f a; a.x = s_lo2[arow + k0]; a.y = s_lo2[arow + k0 + 1];
      acc = __builtin_amdgcn_wmma_f32_16x16x4_f32(false, a, false, b2[c],
                                                  (short)0, acc, false, false);
    }
    const int nb = n0 + 8 * half;
    float* dst = (j == 0) ? lo3b : (j == 1) ? yh2a : yh2b;
    if (j < 3) {
      nt_store4(dst + nb,     acc[0], acc[1], acc[2], acc[3]);
      nt_store4(dst + nb + 4, acc[4], acc[5], acc[6], acc[7]);
    }
  }
}

extern "C" void kernel_launch(void* const* d_in, const int* in_sizes, int n_in,
                              void* d_out, int out_size, void* d_ws, size_t ws_size,
                              hipStream_t stream) {
  (void)in_sizes; (void)n_in; (void)out_size; (void)d_ws; (void)ws_size;
  const float* x   = (const float*)d_in[0];
  const float* h0o = (const float*)d_in[1];
  const float* h1o = (const float*)d_in[2];
  const float* h0a = (const float*)d_in[3];
  const float* h1a = (const float*)d_in[4];
  // d_in[5] = h0b: unused by the reference computation
  const float* h1b = (const float*)d_in[6];

  float* out   = (float*)d_out;
  float* o_lo  = out;                              // [64][131072]
  float* o_yh0 = o_lo  + (size_t)B * L3;           // [64][1][1][524288]
  float* o_yh1 = o_yh0 + (size_t)B * L1;           // [64][2][1][262144]
  float* o_yh2 = o_yh1 + (size_t)B * 2 * L2;       // [64][2][1][131072]

  dtcwt1d_fused<<<dim3(B * GPB), 256, 0, stream>>>(
      x, h0o, h1o, h0a, h1a, h1b, o_lo, o_yh0, o_yh1, o_yh2);
}